// MaximalCodingRateReduction_13606456394205
// MI455X (gfx1250) — compile-verified
//
#include <hip/hip_runtime.h>
#include <hip/hip_bf16.h>

typedef float v2f __attribute__((ext_vector_type(2)));
typedef float v8f __attribute__((ext_vector_type(8)));

#define GAM1f 1.0f
#define GAM2f 1.0f
#define EPSV  0.01f
#define NCLS  10
#define PDIM  64
#define CHUNK 256
#define LDSW  72   // padded LDS row stride (floats): 2*72 mod 64 == 16 -> halves hit disjoint banks

// Dynamic LDS layout (floats): [chunk tile][cls][ord][cstart(16)][acc: NCLS x 64 x 64]
#define SMEM_FLOATS (CHUNK * LDSW + 2 * CHUNK + 16 + NCLS * PDIM * PDIM)

// In-place accumulate: D == C tied to the same (even-aligned) VGPR tuple.
__device__ __forceinline__ void wmma_acc(v8f& acc, v2f a, v2f b) {
    asm volatile("v_wmma_f32_16x16x4_f32 %0, %1, %2, %0"
                 : "+v"(acc)
                 : "v"(a), "v"(b));
}

// ---------------------------------------------------------------------------
// Kernel 1: streaming per-class Gram accumulation with V_WMMA_F32_16X16X4_F32.
// Per chunk: stable rank-sort by class (indices only, deterministic), then
// iterate class segments [cstart[c], cstart[c+1]). K=4 groups are carved from
// each segment's own base, so EVERY group is single-class by construction:
// the inner loop is load -> in-place WMMA on a class-local accumulator, with
// no cross-group state, no masking (except the <=1 partial tail group per
// segment), and no register-array indexing (per-class accumulators live in
// LDS, addressed by class id). 512 thr = 16 waves = all 16 (i,j) 16x16 tiles
// of the 64x64 Gram; partials written per-workgroup to d_ws (no atomics).
// ---------------------------------------------------------------------------
__global__ __launch_bounds__(512) void mcr_gram_kernel(
    const float* __restrict__ embed, const int* __restrict__ targets,
    int m, float* __restrict__ partials /* [nwg][NCLS][64][64] */)
{
    extern __shared__ float smem[];
    float* tile   = smem;                                  // CHUNK*LDSW
    int*   cls    = (int*)(smem + CHUNK * LDSW);           // CHUNK
    int*   ord    = cls + CHUNK;                           // CHUNK
    int*   cstart = ord + CHUNK;                           // 16 (uses 0..NCLS)
    float* accL   = (float*)(cstart + 16);                 // [NCLS][64*64]

    const int tid  = threadIdx.x;
    const int lane = tid & 31;
    const int wv   = tid >> 5;              // 0..15
    const int ti   = (wv >> 2) * 16;        // A row-tile base (p index)
    const int tj   = (wv & 3) * 16;         // B col-tile base (p index)
    const int half = lane >> 4;             // 0: lanes 0-15, 1: lanes 16-31
    const int r    = lane & 15;
    const int colb = tj + r;                // this lane's Gram column

    // Zero the LDS class accumulators.
    for (int e = tid; e < NCLS * PDIM * PDIM; e += 512) accL[e] = 0.0f;

    const int nchunk = m / CHUNK;
    for (int ch = blockIdx.x; ch < nchunk; ch += gridDim.x) {
        const long base = (long)ch * CHUNK;

        // Cooperative, coalesced load of CHUNK x 64 fp32 into padded LDS.
        for (int idx = tid; idx < CHUNK * 16; idx += 512) {
            const int row = idx >> 4, c4 = idx & 15;
            const float4 v = *(const float4*)(embed + (base + row) * PDIM + c4 * 4);
            float* dst = &tile[row * LDSW + c4 * 4];
            dst[0] = v.x; dst[1] = v.y; dst[2] = v.z; dst[3] = v.w;
        }
        if (tid < CHUNK) cls[tid] = targets[base + tid];
        __syncthreads();

        // Stable rank sort by class (indices only) + class-start prefix.
        if (tid < CHUNK) {
            const int c = cls[tid];
            int rank = 0;
            for (int u = 0; u < CHUNK; ++u) {
                const int cu = cls[u];           // broadcast read, no conflicts
                rank += (cu < c) | ((cu == c) & (u < tid));
            }
            ord[rank] = tid;
        }
        if (tid <= NCLS) {
            int cnt = 0;
            for (int u = 0; u < CHUNK; ++u) cnt += (cls[u] < tid);
            cstart[tid] = cnt;                   // cstart[0]=0, cstart[NCLS]=CHUNK
        }
        __syncthreads();

        // Per-class segments; every K=4 group is single-class by construction.
        for (int c = 0; c < NCLS; ++c) {
            const int s0e = __builtin_amdgcn_readfirstlane(cstart[c]);
            const int s1e = __builtin_amdgcn_readfirstlane(cstart[c + 1]);
            const int n = s1e - s0e;
            if (n <= 0) continue;

            v8f tAcc;
            #pragma unroll
            for (int v = 0; v < 8; ++v) tAcc[v] = 0.0f;

            int pos = s0e;
            const int nfull = n >> 2;
            for (int g = 0; g < nfull; ++g) {
                // WMMA f32 16x16x4 layout: lane<16 -> K slots 0,1; lane>=16 -> 2,3.
                const int r0 = ord[pos + half * 2 + 0];
                const int r1 = ord[pos + half * 2 + 1];
                v2f a, b;
                a[0] = tile[r0 * LDSW + ti + r];
                a[1] = tile[r1 * LDSW + ti + r];
                b[0] = tile[r0 * LDSW + tj + r];
                b[1] = tile[r1 * LDSW + tj + r];
                wmma_acc(tAcc, a, b);            // HOT: one in-place WMMA / group
                pos += 4;
            }
            if (n & 3) {
                // Partial tail group: zero invalid B rows (A x 0 == 0).
                const int i0 = pos + half * 2 + 0;
                const int i1 = i0 + 1;
                const int k0 = (i0 < CHUNK) ? i0 : (CHUNK - 1);
                const int k1 = (i1 < CHUNK) ? i1 : (CHUNK - 1);
                const int r0 = ord[k0];
                const int r1 = ord[k1];
                v2f a, b;
                a[0] = tile[r0 * LDSW + ti + r];
                a[1] = tile[r1 * LDSW + ti + r];
                b[0] = (i0 < s1e) ? tile[r0 * LDSW + tj + r] : 0.0f;
                b[1] = (i1 < s1e) ? tile[r1 * LDSW + tj + r] : 0.0f;
                wmma_acc(tAcc, a, b);
            }

            // Flush tAcc into accL[c] (runtime class id = pure address math).
            float* p = accL + (c << 12);
            #pragma unroll
            for (int v = 0; v < 8; ++v) {
                const int row = ti + (half ? v + 8 : v);
                p[row * PDIM + colb] += tAcc[v];
            }
        }
        __syncthreads();
    }

    // Coalesced copy-out of the per-workgroup class Grams.
    float* out = partials + (size_t)blockIdx.x * (NCLS * PDIM * PDIM);
    for (int e = tid; e < NCLS * PDIM * PDIM; e += 512) out[e] = accL[e];
}

// ---------------------------------------------------------------------------
// Kernel 2: class counts + fixed-order reduction of partials, and build the
// 11 SPD matrices M_c = I + scale_c * G_c  (index 10 = total Gram).
// ---------------------------------------------------------------------------
__global__ __launch_bounds__(512) void mcr_reduce_kernel(
    const int* __restrict__ targets, int m,
    const float* __restrict__ partials, int nwg,
    float* __restrict__ mats /* [11][4096] */, float* __restrict__ cnts /* [NCLS] */)
{
    __shared__ float hist[512 * NCLS];
    __shared__ float cntS[NCLS];
    const int tid = threadIdx.x;

    for (int c = 0; c < NCLS; ++c) hist[tid * NCLS + c] = 0.0f;
    for (int idx = tid; idx < m; idx += 512) {
        const int t = targets[idx];
        hist[tid * NCLS + t] += 1.0f;
    }
    __syncthreads();
    if (tid < NCLS) {
        float s = 0.0f;
        for (int t = 0; t < 512; ++t) s += hist[t * NCLS + tid];
        cntS[tid] = s;
        cnts[tid] = s;
    }
    __syncthreads();

    const float sct = GAM1f * (float)PDIM / ((float)m * EPSV);
    for (int e = tid; e < PDIM * PDIM; e += 512) {
        const int row = e >> 6, col = e & 63;
        const float diag = (row == col) ? 1.0f : 0.0f;
        float tot = 0.0f;
        for (int c = 0; c < NCLS; ++c) {
            float g = 0.0f;
            for (int w = 0; w < nwg; ++w)
                g += partials[(size_t)w * (NCLS * PDIM * PDIM) + c * PDIM * PDIM + e];
            tot += g;
            const float trPi = cntS[c] + 1e-8f;
            const float sc = (float)PDIM / (trPi * EPSV);
            mats[c * PDIM * PDIM + e] = diag + sc * g;
        }
        mats[NCLS * PDIM * PDIM + e] = diag + sct * tot;
    }
}

// ---------------------------------------------------------------------------
// Kernel 3: 64x64 Cholesky logdet; one block per matrix (grid = 11).
// ---------------------------------------------------------------------------
__global__ __launch_bounds__(256) void mcr_chol_kernel(
    const float* __restrict__ mats, float* __restrict__ logdets /* [11] */)
{
    __shared__ float A[PDIM * 65];   // padded
    const int tid = threadIdx.x;
    const int b = blockIdx.x;

    for (int e = tid; e < PDIM * PDIM; e += 256)
        A[(e >> 6) * 65 + (e & 63)] = mats[b * PDIM * PDIM + e];
    __syncthreads();

    float ldacc = 0.0f;
    for (int k = 0; k < PDIM; ++k) {
        const float d = sqrtf(A[k * 65 + k]);
        const int i = k + 1 + tid;
        if (i < PDIM) A[i * 65 + k] /= d;
        __syncthreads();
        if (i < PDIM) {
            const float lik = A[i * 65 + k];
            for (int j = k + 1; j <= i; ++j)
                A[i * 65 + j] -= lik * A[j * 65 + k];
        }
        if (tid == 0) ldacc += 2.0f * logf(d);
        __syncthreads();
    }
    if (tid == 0) logdets[b] = ldacc;
}

// ---------------------------------------------------------------------------
// Kernel 4: final scalar combine.
// ---------------------------------------------------------------------------
__global__ void mcr_final_kernel(const float* __restrict__ logdets,
                                 const float* __restrict__ cnts,
                                 int m, float* __restrict__ out)
{
    if (threadIdx.x == 0 && blockIdx.x == 0) {
        const float discrimn = 0.5f * logdets[NCLS];
        float compress = 0.0f;
        for (int c = 0; c < NCLS; ++c) {
            const float trPi = cnts[c] + 1e-8f;
            compress += logdets[c] * trPi / (float)m;
        }
        compress *= 0.5f;
        out[0] = GAM2f * (-discrimn) + compress;
    }
}

extern "C" void kernel_launch(void* const* d_in, const int* in_sizes, int n_in,
                              void* d_out, int out_size, void* d_ws, size_t ws_size,
                              hipStream_t stream) {
    (void)n_in; (void)out_size;
    const float* embed   = (const float*)d_in[0];
    const int*   targets = (const int*)d_in[1];
    const int m = in_sizes[1];                       // number of samples

    // Workspace partition (floats): [nwg partial Grams][11 mats][counts][logdets]
    const size_t per_wg  = (size_t)NCLS * PDIM * PDIM;
    const size_t reserve = 11 * PDIM * PDIM + 32;
    const size_t avail   = ws_size / sizeof(float);
    int nwg = 64;
    if (avail > reserve) {
        const size_t maxwg = (avail - reserve) / per_wg;
        if ((size_t)nwg > maxwg) nwg = (int)maxwg;
    }
    if (nwg < 1) nwg = 1;

    float* partials = (float*)d_ws;
    float* mats     = partials + (size_t)nwg * per_wg;
    float* cnts     = mats + 11 * PDIM * PDIM;
    float* lgd      = cnts + 16;

    const size_t smem_bytes = (size_t)SMEM_FLOATS * sizeof(float);
    hipFuncSetAttribute((const void*)mcr_gram_kernel,
                        hipFuncAttributeMaxDynamicSharedMemorySize,
                        (int)smem_bytes);

    mcr_gram_kernel  <<<nwg, 512, smem_bytes, stream>>>(embed, targets, m, partials);
    mcr_reduce_kernel<<<1,   512, 0, stream>>>(targets, m, partials, nwg, mats, cnts);
    mcr_chol_kernel  <<<11,  256, 0, stream>>>(mats, lgd);
    mcr_final_kernel <<<1,   64,  0, stream>>>(lgd, cnts, m, (float*)d_out);
}